// HyenaLayer_64398739636494
// MI455X (gfx1250) — compile-verified
//
#include <hip/hip_runtime.h>
#include <cstddef>
#include <cstdint>

#define THREADS 256
#define BM 64
#define BN 64
#define BK 32
#define KSTR 36          // padded LDS k-stride: 16B-aligned rows, conflict-free fragments
#define NCHUNK (512 / BK)
#define FFT_N 4096
#define FFT_NH 2048
#define TDIM 2048
#define WDIM 512
#define BDIM 4
#define NSTAGE 3

typedef __attribute__((ext_vector_type(2))) float v2f;
typedef __attribute__((ext_vector_type(8))) float v8f;

// ---------------- CDNA5 async global->LDS copies (ASYNCcnt-tracked) ----------------
__device__ __forceinline__ uint32_t lds_off(const void* p) {
  // generic pointer to __shared__ = {shared_aperture_hi, lds_byte_offset}; low 32 bits
  // are exactly the per-wave LDS offset the VGLOBAL async ops consume (ISA 10.2).
  return (uint32_t)(uintptr_t)p;
}
__device__ __forceinline__ void async_b32(uint32_t dst_lds, const void* gsrc) {
  asm volatile("global_load_async_to_lds_b32 %0, %1, off"
               :: "v"(dst_lds), "v"(gsrc) : "memory");
}
__device__ __forceinline__ void async_b128(uint32_t dst_lds, const void* gsrc) {
  asm volatile("global_load_async_to_lds_b128 %0, %1, off"
               :: "v"(dst_lds), "v"(gsrc) : "memory");
}
__device__ __forceinline__ void wait_async0() {
  asm volatile("s_wait_asynccnt 0x0" ::: "memory");
}

// ---------------- WMMA core: As[m][k], Bs[n][k], wave computes 16x32 ----------------
__device__ __forceinline__ void wmma_chunk(const float* As, const float* Bs,
                                           int wr, int wc, int lane,
                                           v8f& c0, v8f& c1) {
  const int l16 = lane & 15;
  const int kh  = (lane >> 4) << 1;   // lanes 0-15: K=k4+0,1 ; lanes 16-31: K=k4+2,3
#pragma unroll
  for (int k4 = 0; k4 < BK; k4 += 4) {
    v2f a  = *(const v2f*)&As[(wr * 16 + l16) * KSTR + k4 + kh];
    v2f b0 = *(const v2f*)&Bs[(wc * 32 + l16) * KSTR + k4 + kh];
    v2f b1 = *(const v2f*)&Bs[(wc * 32 + 16 + l16) * KSTR + k4 + kh];
    c0 = __builtin_amdgcn_wmma_f32_16x16x4_f32(false, a, false, b0, (short)0, c0, false, false);
    c1 = __builtin_amdgcn_wmma_f32_16x16x4_f32(false, a, false, b1, (short)0, c1, false, false);
  }
}

__device__ __forceinline__ void store_tile(float* p, const v8f& c) {
  float4 lo = make_float4(c[0], c[1], c[2], c[3]);
  float4 hi = make_float4(c[4], c[5], c[6], c[7]);
  *(float4*)p       = lo;
  *(float4*)(p + 4) = hi;
}

// ---------------- GEMM 1: XP[s][b][v][t] = (X @ W_proj[:,:,s]) transposed ----------------
// Double-buffered LDS; tiles fetched with async global->LDS, overlapped with WMMA.
__global__ __launch_bounds__(THREADS) void gemm_proj_kernel(
    const float* __restrict__ X,    // [B*T=8192][512]
    const float* __restrict__ Wp,   // [512][512][4]
    float* __restrict__ XP)         // [4][4][512][2048]
{
  __shared__ __align__(16) float As[2][BM * KSTR];
  __shared__ __align__(16) float Bs[2][BN * KSTR];
  const int tid  = threadIdx.x;
  const int s    = blockIdx.z;
  const int tm   = blockIdx.y * BM;
  const int tn   = blockIdx.x * BN;
  const int wave = tid >> 5, lane = tid & 31;
  const int wr   = wave & 3, wc = wave >> 2;
  v8f c0 = {0.f, 0.f, 0.f, 0.f, 0.f, 0.f, 0.f, 0.f};
  v8f c1 = {0.f, 0.f, 0.f, 0.f, 0.f, 0.f, 0.f, 0.f};

  const int rA = tid >> 3;           // A rows rA and rA+32
  const int cA = (tid & 7) << 2;     // A col group (float4)

  auto issue_tiles = [&](int buf, int kk) {
    // A tile 64x32: two 16B async copies per thread
    async_b128(lds_off(&As[buf][rA * KSTR + cA]),
               X + (size_t)(tm + rA) * WDIM + kk + cA);
    async_b128(lds_off(&As[buf][(rA + 32) * KSTR + cA]),
               X + (size_t)(tm + rA + 32) * WDIM + kk + cA);
    // B tile 32x64 (stride-4 gather) -> Bs[n][k]: eight 4B async copies per thread
#pragma unroll
    for (int it = 0; it < 8; ++it) {
      int j = tid + it * THREADS;
      int k = j >> 6, n = j & 63;
      async_b32(lds_off(&Bs[buf][n * KSTR + k]),
                Wp + ((size_t)(kk + k) * WDIM + (tn + n)) * 4 + s);
    }
  };

  issue_tiles(0, 0);
  wait_async0();
  __syncthreads();
  for (int ci = 0; ci < NCHUNK; ++ci) {
    const int cur = ci & 1;
    if (ci + 1 < NCHUNK) issue_tiles(cur ^ 1, (ci + 1) * BK);   // prefetch next chunk
    wmma_chunk(As[cur], Bs[cur], wr, wc, lane, c0, c1);         // compute current
    wait_async0();
    __syncthreads();
  }

  const int l16 = lane & 15;
  const int mo  = wr * 16 + ((lane >> 4) << 3);
  const int m0  = tm + mo;
  const int b   = m0 >> 11;
  const int t0  = m0 & (TDIM - 1);
  float* base = XP + ((size_t)(s * BDIM + b) * WDIM) * TDIM;
  store_tile(base + (size_t)(tn + wc * 32 + l16) * TDIM + t0, c0);
  store_tile(base + (size_t)(tn + wc * 32 + 16 + l16) * TDIM + t0, c1);
}

// ---------------- GEMM 2: F[s][v][t] = (pe @ W_filt[:,:,s]) transposed; pe on the fly ----
__global__ __launch_bounds__(THREADS) void gemm_filt_kernel(
    const float* __restrict__ Wf,   // [512][512][3]
    float* __restrict__ F)          // [3][512][2048]
{
  __shared__ __align__(16) float As[2][BM * KSTR];
  __shared__ __align__(16) float Bs[2][BN * KSTR];
  const int tid  = threadIdx.x;
  const int s    = blockIdx.z;
  const int tm   = blockIdx.y * BM;
  const int tn   = blockIdx.x * BN;
  const int wave = tid >> 5, lane = tid & 31;
  const int wr   = wave & 3, wc = wave >> 2;
  v8f c0 = {0.f, 0.f, 0.f, 0.f, 0.f, 0.f, 0.f, 0.f};
  v8f c1 = {0.f, 0.f, 0.f, 0.f, 0.f, 0.f, 0.f, 0.f};
  const float LN1E4_O_HALF = 9.2103403719761836f / 256.0f;

  auto fill_A = [&](int buf, int kk) {   // positional encoding computed in-place
#pragma unroll
    for (int it = 0; it < 2; ++it) {
      int i = tid + it * THREADS;
      int r = i >> 3, c = (i & 7) << 2;
      int t = tm + r;
#pragma unroll
      for (int u = 0; u < 4; ++u) {
        int w = kk + c + u;
        int j = w & 255;
        float fr  = __expf(-(float)j * LN1E4_O_HALF);
        float ang = (float)t * fr;
        As[buf][r * KSTR + c + u] = (w < 256) ? sinf(ang) : cosf(ang);
      }
    }
  };
  auto issue_B = [&](int buf, int kk) {
#pragma unroll
    for (int it = 0; it < 8; ++it) {
      int j = tid + it * THREADS;
      int k = j >> 6, n = j & 63;
      async_b32(lds_off(&Bs[buf][n * KSTR + k]),
                Wf + ((size_t)(kk + k) * WDIM + (tn + n)) * 3 + s);
    }
  };

  issue_B(0, 0);
  fill_A(0, 0);
  wait_async0();
  __syncthreads();
  for (int ci = 0; ci < NCHUNK; ++ci) {
    const int cur = ci & 1;
    if (ci + 1 < NCHUNK) { issue_B(cur ^ 1, (ci + 1) * BK); fill_A(cur ^ 1, (ci + 1) * BK); }
    wmma_chunk(As[cur], Bs[cur], wr, wc, lane, c0, c1);
    wait_async0();
    __syncthreads();
  }

  const int l16 = lane & 15;
  const int mo  = wr * 16 + ((lane >> 4) << 3);
  const int t0  = tm + mo;
  float* base = F + (size_t)(s * WDIM) * TDIM;
  store_tile(base + (size_t)(tn + wc * 32 + l16) * TDIM + t0, c0);
  store_tile(base + (size_t)(tn + wc * 32 + 16 + l16) * TDIM + t0, c1);
}

// ---------------- LDS Stockham radix-2 FFT, N=4096 ----------------
__device__ __forceinline__ void fft_fill_twiddles(float2* tw, int tid) {
  for (int k = tid; k < FFT_NH; k += THREADS) {
    float ang = -6.28318530717958647692f * ((float)k / (float)FFT_N);
    float sv, cv;
    sincosf(ang, &sv, &cv);
    tw[k] = make_float2(cv, sv);          // exp(-2*pi*i*k/N)
  }
}

// isign = +1: forward, -1: inverse (unscaled). Ends with result in A (12 passes).
__device__ __forceinline__ void fft4096(float2* A, float2* B, const float2* tw,
                                        float isign, int tid) {
  float2* src = A;
  float2* dst = B;
  for (int s = 1; s < FFT_N; s <<= 1) {
    for (int j = tid; j < FFT_NH; j += THREADS) {
      int q  = j & (s - 1);
      int ps = j - q;                      // p*s == twiddle index
      float2 a = src[j];
      float2 b = src[j + FFT_NH];
      float2 w = tw[ps];
      float wi = w.y * isign;
      float dr = a.x - b.x;
      float di = a.y - b.y;
      dst[j + ps]     = make_float2(a.x + b.x, a.y + b.y);
      dst[j + ps + s] = make_float2(dr * w.x - di * wi, dr * wi + di * w.x);
    }
    __syncthreads();
    float2* t = src; src = dst; dst = t;
  }
}

// ---------------- Filter spectra: FF[s][v][k] = FFT_4096(zeropad(F[s][v][:])) ----------
__global__ __launch_bounds__(THREADS) void filter_fft_kernel(
    const float* __restrict__ F,    // [3][512][2048]
    float* __restrict__ FF)         // [3][512][4096][2]
{
  __shared__ float2 bufA[FFT_N];
  __shared__ float2 bufB[FFT_N];
  __shared__ float2 tw[FFT_NH];
  const int tid = threadIdx.x;
  const int ch  = blockIdx.x;       // s*512 + v
  fft_fill_twiddles(tw, tid);
  const float* f = F + (size_t)ch * TDIM;
  for (int t = tid; t < FFT_N; t += THREADS)
    bufA[t] = (t < TDIM) ? make_float2(f[t], 0.f) : make_float2(0.f, 0.f);
  __syncthreads();
  fft4096(bufA, bufB, tw, 1.0f, tid);
  float2* out = (float2*)FF + (size_t)ch * FFT_N;
  for (int k = tid; k < FFT_N; k += THREADS) out[k] = bufA[k];
}

// ---------------- 3-stage FFT convolution + gating, one channel per workgroup --------
__global__ __launch_bounds__(THREADS) void conv_stages_kernel(
    const float* __restrict__ XP,   // [4][4][512][2048]
    const float* __restrict__ FF,   // [3][512][4096][2]
    float* __restrict__ out)        // [4][2048][512]
{
  __shared__ float2 bufA[FFT_N];
  __shared__ float2 bufB[FFT_N];
  __shared__ float2 tw[FFT_NH];
  const int tid = threadIdx.x;
  const int ch  = blockIdx.x;       // b*512 + v
  const int b   = ch >> 9;
  const int v   = ch & 511;
  fft_fill_twiddles(tw, tid);

  const float* y0 = XP + ((size_t)(0 * BDIM + b) * WDIM + v) * TDIM;
  for (int t = tid; t < FFT_N; t += THREADS)
    bufA[t] = (t < TDIM) ? make_float2(y0[t], 0.f) : make_float2(0.f, 0.f);
  __syncthreads();

  const float invn = 1.0f / (float)FFT_N;
  for (int s = 0; s < NSTAGE; ++s) {
    fft4096(bufA, bufB, tw, 1.0f, tid);                 // forward, result in bufA
    const float2* ff = (const float2*)FF + ((size_t)s * WDIM + v) * FFT_N;
    for (int k = tid; k < FFT_N; k += THREADS) {
      float2 y = bufA[k];
      float2 f = ff[k];
      bufA[k] = make_float2(y.x * f.x - y.y * f.y, y.x * f.y + y.y * f.x);
    }
    __syncthreads();
    fft4096(bufA, bufB, tw, -1.0f, tid);                // inverse (unscaled)
    const float* g = XP + ((size_t)((s + 1) * BDIM + b) * WDIM + v) * TDIM;
    for (int t = tid; t < FFT_N; t += THREADS) {
      float val = (t < TDIM) ? bufA[t].x * invn * g[t] : 0.f;   // gate + causal re-zero
      bufA[t] = make_float2(val, 0.f);
    }
    __syncthreads();
  }
  for (int t = tid; t < TDIM; t += THREADS)
    out[((size_t)b * TDIM + t) * WDIM + v] = bufA[t].x;
}

// ---------------- launch ----------------
extern "C" void kernel_launch(void* const* d_in, const int* in_sizes, int n_in,
                              void* d_out, int out_size, void* d_ws, size_t ws_size,
                              hipStream_t stream) {
  const float* X  = (const float*)d_in[0];   // (4,2048,512)
  const float* Wp = (const float*)d_in[1];   // (512,512,4)
  const float* Wf = (const float*)d_in[2];   // (512,512,3)
  float* out = (float*)d_out;
  float* ws  = (float*)d_ws;

  // workspace layout (floats)
  const size_t XP_OFF = 0;                                        // 16,777,216
  const size_t F_OFF  = XP_OFF + (size_t)4 * BDIM * WDIM * TDIM;  // + 3,145,728
  const size_t FF_OFF = F_OFF + (size_t)NSTAGE * WDIM * TDIM;     // + 12,582,912
  float* XP = ws + XP_OFF;
  float* F  = ws + F_OFF;
  float* FF = ws + FF_OFF;

  dim3 blk(THREADS);
  gemm_proj_kernel<<<dim3(WDIM / BN, (BDIM * TDIM) / BM, 4), blk, 0, stream>>>(X, Wp, XP);
  gemm_filt_kernel<<<dim3(WDIM / BN, TDIM / BM, NSTAGE), blk, 0, stream>>>(Wf, F);
  filter_fft_kernel<<<dim3(NSTAGE * WDIM), blk, 0, stream>>>(F, FF);
  conv_stages_kernel<<<dim3(BDIM * WDIM), blk, 0, stream>>>(XP, FF, out);
}